// Rs_GCN_23347442221529
// MI455X (gfx1250) — compile-verified
//
#include <hip/hip_runtime.h>
#include <math.h>

// Problem constants (from reference): B=192, F=256, N=512, top-K=10
#define BB 192
#define FF 256
#define NN 512
#define K_TOP 10

typedef __attribute__((ext_vector_type(2))) float v2f;
typedef __attribute__((ext_vector_type(8))) float v8f;
typedef __attribute__((ext_vector_type(4))) unsigned int v4u;
typedef __attribute__((ext_vector_type(8))) int v8i;
typedef __attribute__((ext_vector_type(4))) int v4i;

#if defined(__has_builtin)
#if __has_builtin(__builtin_amdgcn_tensor_load_to_lds)
#define HAVE_TDM 1
#endif
#endif
#ifndef HAVE_TDM
#define HAVE_TDM 0
#endif

// D = A(16x4 f32) * B(4x16 f32) + C(16x16 f32)  -- CDNA5 fp32 WMMA
__device__ __forceinline__ v8f wmma_f32(v2f a, v2f b, v8f c) {
    return __builtin_amdgcn_wmma_f32_16x16x4_f32(
        /*neg_a=*/false, a, /*neg_b=*/false, b,
        /*c_mod=*/(short)0, c, /*reuse_a=*/false, /*reuse_b=*/false);
}

__device__ __forceinline__ float wave_max32(float x) {
    #pragma unroll
    for (int off = 16; off > 0; off >>= 1)
        x = fmaxf(x, __shfl_xor(x, off, 32));
    return x;
}
__device__ __forceinline__ float wave_sum32(float x) {
    #pragma unroll
    for (int off = 16; off > 0; off >>= 1)
        x += __shfl_xor(x, off, 32);
    return x;
}

// ---------------------------------------------------------------------------
// Stage a K x 16 float column-panel (row stride rowStride floats) into LDS
// using the Tensor Data Mover (one wave issues; TENSORcnt + barrier sync).
// D# built per CDNA5 ISA 8.3/8.4: 2D tile, data_size=4B, tile_dim0=16.
// ---------------------------------------------------------------------------
__device__ __forceinline__ void stage_panel16(const float* __restrict__ gsrc,
                                              float* lds_ptr, int K, int rowStride) {
#if HAVE_TDM
    if (threadIdx.x < 32) {
        unsigned long long ga = (unsigned long long)(uintptr_t)gsrc;
        unsigned lds_off = (unsigned)(uintptr_t)lds_ptr;   // LDS byte offset (addr[31:0])
        v4u g0;
        g0[0] = 1u;                                        // count=1 (valid descriptor)
        g0[1] = lds_off;                                   // lds_addr (bytes)
        g0[2] = (unsigned)(ga & 0xFFFFFFFFull);            // global_addr[31:0]
        g0[3] = (unsigned)((ga >> 32) & 0x1FFFFFFull)      // global_addr[56:32]
              | (2u << 30);                                // type=2 ("image")
        v8i g1;
        g1[0] = (int)(2u << 16);                           // data_size=2 -> 4 bytes
        g1[1] = (int)(16u << 16);                          // tensor_dim0=16 (bits 63:48)
        g1[2] = (int)(((unsigned)K) << 16);                // tensor_dim1=K  (bits 95:80)
        g1[3] = (int)(16u << 16);                          // tile_dim0=16   (bits 127:112)
        g1[4] = (int)(unsigned)K;                          // tile_dim1=K    (bits 143:128)
        g1[5] = rowStride;                                 // tensor_dim0_stride[31:0]
        g1[6] = 0;
        g1[7] = 0;
        v4i z4 = {};
#if __clang_major__ >= 23
        v8i z8 = {};
        __builtin_amdgcn_tensor_load_to_lds(g0, g1, z4, z4, z8, 0);
#else
        __builtin_amdgcn_tensor_load_to_lds(g0, g1, z4, z4, 0);
#endif
        __builtin_amdgcn_s_wait_tensorcnt(0);
    }
    __syncthreads();
#else
    for (int i = threadIdx.x; i < K * 16; i += blockDim.x)
        lds_ptr[i] = gsrc[(i >> 4) * rowStride + (i & 15)];
    __syncthreads();
#endif
}

// ---------------------------------------------------------------------------
// prep: M1[f][h] = sum_k g_w[k][f] * w1[k][h]   (g_w^T @ w1)
//       c1[h]    = sum_k g_b[k]    * w1[k][h]
// ---------------------------------------------------------------------------
__global__ __launch_bounds__(FF) void prep_kernel(const float* __restrict__ g_w,
                                                  const float* __restrict__ w1,
                                                  const float* __restrict__ g_b,
                                                  float* __restrict__ M1,
                                                  float* __restrict__ c1) {
    int h = threadIdx.x;
    int f = blockIdx.x;
    float s = 0.f;
    for (int k = 0; k < FF; ++k)
        s = fmaf(g_w[k * FF + f], w1[k * FF + h], s);
    M1[f * FF + h] = s;
    if (f == 0) {
        float t = 0.f;
        for (int k = 0; k < FF; ++k)
            t = fmaf(g_b[k], w1[k * FF + h], t);
        c1[h] = t;
    }
}

// ---------------------------------------------------------------------------
// gram: A[b][n][m] = sum_f v[b][f][n] * v[b][f][m]
// Block = 8 waves: shared A-panel v[:, n0:n0+16] staged in LDS via TDM;
// wave w computes the 16x16 tile at (n0, gy*128 + w*16).
// ---------------------------------------------------------------------------
__global__ __launch_bounds__(256) void gram_kernel(const float* __restrict__ v,
                                                   float* __restrict__ A) {
    __shared__ float ldsA[FF * 16];                 // 16 KB
    const int n0 = blockIdx.x * 16, b = blockIdx.z;
    const int wv = threadIdx.x >> 5, lane = threadIdx.x & 31;
    const int m0 = blockIdx.y * 128 + wv * 16;
    const int half = lane >> 4, l = lane & 15, kb = 2 * half;
    const float* vb = v + (size_t)b * FF * NN;

    stage_panel16(vb + n0, ldsA, FF, NN);

    v8f acc = {};
    for (int f0 = 0; f0 < FF; f0 += 4) {
        v2f a, bm;
        a.x  = ldsA[(f0 + kb) * 16 + l];
        a.y  = ldsA[(f0 + kb + 1) * 16 + l];
        bm.x = vb[(size_t)(f0 + kb)     * NN + m0 + l];
        bm.y = vb[(size_t)(f0 + kb + 1) * NN + m0 + l];
        acc = wmma_f32(a, bm, acc);
    }
    float* Ab = A + (size_t)b * NN * NN;
    #pragma unroll
    for (int r = 0; r < 8; ++r)
        Ab[(size_t)(n0 + r + 8 * half) * NN + m0 + l] = acc[r];
}

// ---------------------------------------------------------------------------
// topk: keep exactly top-10 per row, add self loop, compute d = deg^-1/2
// (inf -> 0 matching reference). One wave per row, ballot-based consumption.
// ---------------------------------------------------------------------------
__global__ __launch_bounds__(32) void topk_kernel(float* __restrict__ A,
                                                  float* __restrict__ dvec) {
    const int n = blockIdx.x, b = blockIdx.y;
    const int lane = threadIdx.x;
    float* row = A + ((size_t)b * NN + n) * NN;

    float orig[16], work[16];
    #pragma unroll
    for (int j = 0; j < 16; ++j) {
        orig[j] = row[lane + 32 * j];
        work[j] = orig[j];
    }

    unsigned keep = 0u;
    for (int t = 0; t < K_TOP; ++t) {
        float lm = -INFINITY; int ls = 0;
        #pragma unroll
        for (int j = 0; j < 16; ++j)
            if (work[j] > lm) { lm = work[j]; ls = j; }
        float gm = wave_max32(lm);
        unsigned long long ball = __ballot(lm == gm);
        int first = (int)__ffsll(ball) - 1;
        if (lane == first) {
            work[ls] = -INFINITY;
            keep |= (1u << ls);
        }
    }

    float rs = 0.f;
    #pragma unroll
    for (int j = 0; j < 16; ++j) {
        int m = lane + 32 * j;
        float x = ((keep >> j) & 1u) ? orig[j] : 0.f;
        if (m == n) x += 1.0f;
        rs += x;
        row[m] = x;
    }
    rs = wave_sum32(rs);
    if (lane == 0) {
        float d = rsqrtf(rs);
        if (isinf(d)) d = 0.f;
        dvec[(size_t)b * NN + n] = d;
    }
}

// ---------------------------------------------------------------------------
// h1: h1p[b][m][h] = d[m] * ( sum_f v[b][f][m]*M1[f][h] + c1[h] )
// Shared B-panel M1[:, h0:h0+16] staged via TDM; waves cover 8 m-tiles.
// ---------------------------------------------------------------------------
__global__ __launch_bounds__(256) void h1_kernel(const float* __restrict__ v,
                                                 const float* __restrict__ M1,
                                                 const float* __restrict__ c1,
                                                 const float* __restrict__ dvec,
                                                 float* __restrict__ h1p) {
    __shared__ float ldsB[FF * 16];                 // 16 KB
    const int h0 = blockIdx.x * 16, b = blockIdx.z;
    const int wv = threadIdx.x >> 5, lane = threadIdx.x & 31;
    const int m0 = blockIdx.y * 128 + wv * 16;
    const int half = lane >> 4, l = lane & 15, kb = 2 * half;
    const float* vb = v + (size_t)b * FF * NN;

    stage_panel16(M1 + h0, ldsB, FF, FF);

    v8f acc = {};
    for (int f0 = 0; f0 < FF; f0 += 4) {
        v2f a, bm;
        a.x  = vb[(size_t)(f0 + kb)     * NN + m0 + l];
        a.y  = vb[(size_t)(f0 + kb + 1) * NN + m0 + l];
        bm.x = ldsB[(f0 + kb) * 16 + l];
        bm.y = ldsB[(f0 + kb + 1) * 16 + l];
        acc = wmma_f32(a, bm, acc);
    }
    const float c1v = c1[h0 + l];
    const float* dv = dvec + (size_t)b * NN;
    float* H = h1p + (size_t)b * NN * FF;
    #pragma unroll
    for (int r = 0; r < 8; ++r) {
        int m = m0 + r + 8 * half;
        H[(size_t)m * FF + h0 + l] = dv[m] * (acc[r] + c1v);
    }
}

// ---------------------------------------------------------------------------
// spmm1: X[b][n][h] = relu( d[n] * sum_m A[n][m]*h1p[m][h] + b1[h] )
// Shared B-panel h1p[:, h0:h0+16] (32 KB) via TDM; A rows on coalesced v2f.
// ---------------------------------------------------------------------------
__global__ __launch_bounds__(256) void spmm_relu_kernel(const float* __restrict__ A,
                                                        const float* __restrict__ Hin,
                                                        const float* __restrict__ dvec,
                                                        const float* __restrict__ bias,
                                                        float* __restrict__ X) {
    __shared__ float ldsB[NN * 16];                 // 32 KB
    const int h0 = blockIdx.x * 16, b = blockIdx.z;
    const int wv = threadIdx.x >> 5, lane = threadIdx.x & 31;
    const int n0 = blockIdx.y * 128 + wv * 16;
    const int half = lane >> 4, l = lane & 15, kb = 2 * half;
    const float* Ab = A + (size_t)b * NN * NN;
    const float* Hb = Hin + (size_t)b * NN * FF;

    stage_panel16(Hb + h0, ldsB, NN, FF);

    v8f acc = {};
    for (int m0 = 0; m0 < NN; m0 += 4) {
        v2f a = *(const v2f*)&Ab[(size_t)(n0 + l) * NN + m0 + kb];  // 8B-aligned
        v2f bm;
        bm.x = ldsB[(m0 + kb) * 16 + l];
        bm.y = ldsB[(m0 + kb + 1) * 16 + l];
        acc = wmma_f32(a, bm, acc);
    }
    const float* dv = dvec + (size_t)b * NN;
    const float bv = bias[h0 + l];
    float* Xb = X + (size_t)b * NN * FF;
    #pragma unroll
    for (int r = 0; r < 8; ++r) {
        int n = n0 + r + 8 * half;
        Xb[(size_t)n * FF + h0 + l] = fmaxf(dv[n] * acc[r] + bv, 0.f);
    }
}

// ---------------------------------------------------------------------------
// h2: h2p[b][m][h] = d[m] * sum_f X[b][m][f]*w2[f][h]
// Shared B-panel w2[:, h0:h0+16] via TDM; X rows on coalesced v2f.
// ---------------------------------------------------------------------------
__global__ __launch_bounds__(256) void h2_kernel(const float* __restrict__ X,
                                                 const float* __restrict__ w2,
                                                 const float* __restrict__ dvec,
                                                 float* __restrict__ h2p) {
    __shared__ float ldsB[FF * 16];                 // 16 KB
    const int h0 = blockIdx.x * 16, b = blockIdx.z;
    const int wv = threadIdx.x >> 5, lane = threadIdx.x & 31;
    const int m0 = blockIdx.y * 128 + wv * 16;
    const int half = lane >> 4, l = lane & 15, kb = 2 * half;
    const float* Xb = X + (size_t)b * NN * FF;

    stage_panel16(w2 + h0, ldsB, FF, FF);

    v8f acc = {};
    for (int f0 = 0; f0 < FF; f0 += 4) {
        v2f a = *(const v2f*)&Xb[(size_t)(m0 + l) * FF + f0 + kb];  // 8B-aligned
        v2f bm;
        bm.x = ldsB[(f0 + kb) * 16 + l];
        bm.y = ldsB[(f0 + kb + 1) * 16 + l];
        acc = wmma_f32(a, bm, acc);
    }
    const float* dv = dvec + (size_t)b * NN;
    float* H = h2p + (size_t)b * NN * FF;
    #pragma unroll
    for (int r = 0; r < 8; ++r) {
        int m = m0 + r + 8 * half;
        H[(size_t)m * FF + h0 + l] = dv[m] * acc[r];
    }
}

// ---------------------------------------------------------------------------
// spmm2: out[b][h][n] = d[n]*sum_m A[n][m]*h2p[m][h] + b2[h] + v[b][h][n]
// (transposed store + residual fused)
// ---------------------------------------------------------------------------
__global__ __launch_bounds__(256) void spmm_out_kernel(const float* __restrict__ A,
                                                       const float* __restrict__ Hin,
                                                       const float* __restrict__ dvec,
                                                       const float* __restrict__ bias,
                                                       const float* __restrict__ v,
                                                       float* __restrict__ out) {
    __shared__ float ldsB[NN * 16];                 // 32 KB
    const int h0 = blockIdx.x * 16, b = blockIdx.z;
    const int wv = threadIdx.x >> 5, lane = threadIdx.x & 31;
    const int n0 = blockIdx.y * 128 + wv * 16;
    const int half = lane >> 4, l = lane & 15, kb = 2 * half;
    const float* Ab = A + (size_t)b * NN * NN;
    const float* Hb = Hin + (size_t)b * NN * FF;

    stage_panel16(Hb + h0, ldsB, NN, FF);

    v8f acc = {};
    for (int m0 = 0; m0 < NN; m0 += 4) {
        v2f a = *(const v2f*)&Ab[(size_t)(n0 + l) * NN + m0 + kb];
        v2f bm;
        bm.x = ldsB[(m0 + kb) * 16 + l];
        bm.y = ldsB[(m0 + kb + 1) * 16 + l];
        acc = wmma_f32(a, bm, acc);
    }
    const float* dv = dvec + (size_t)b * NN;
    const float bv = bias[h0 + l];
    const float* vb = v + (size_t)b * FF * NN;
    float* ob = out + (size_t)b * FF * NN;
    const int hh = h0 + l;
    #pragma unroll
    for (int r = 0; r < 8; ++r) {
        int n = n0 + r + 8 * half;
        ob[(size_t)hh * NN + n] = dv[n] * acc[r] + bv + vb[(size_t)hh * NN + n];
    }
}

// ---------------------------------------------------------------------------
extern "C" void kernel_launch(void* const* d_in, const int* in_sizes, int n_in,
                              void* d_out, int out_size, void* d_ws, size_t ws_size,
                              hipStream_t stream) {
    (void)in_sizes; (void)n_in; (void)out_size; (void)ws_size;
    const float* v   = (const float*)d_in[0];
    const float* g_w = (const float*)d_in[1];
    const float* g_b = (const float*)d_in[2];
    const float* w1  = (const float*)d_in[3];
    const float* b1  = (const float*)d_in[4];
    const float* w2  = (const float*)d_in[5];
    const float* b2  = (const float*)d_in[6];
    float* out = (float*)d_out;

    float* ws = (float*)d_ws;
    size_t off = 0;
    float* A    = ws + off; off += (size_t)BB * NN * NN;   // 201 MB
    float* dvec = ws + off; off += (size_t)BB * NN;
    float* h1p  = ws + off; off += (size_t)BB * NN * FF;   // 100 MB (reused as h2p)
    float* x1   = ws + off; off += (size_t)BB * NN * FF;   // 100 MB
    float* M1   = ws + off; off += (size_t)FF * FF;
    float* c1   = ws + off; off += (size_t)FF;

    // weight-fusion prep (tiny)
    prep_kernel<<<dim3(FF), FF, 0, stream>>>(g_w, w1, g_b, M1, c1);
    // A = v^T v
    gram_kernel<<<dim3(NN / 16, NN / 128, BB), 256, 0, stream>>>(v, A);
    // top-10 sparsify + self loop + degree^-1/2
    topk_kernel<<<dim3(NN, BB), 32, 0, stream>>>(A, dvec);
    // h1' = D (v^T (g_w^T w1) + g_b w1)
    h1_kernel<<<dim3(FF / 16, NN / 128, BB), 256, 0, stream>>>(v, M1, c1, dvec, h1p);
    // X1 = relu(D A h1' + b1)
    spmm_relu_kernel<<<dim3(FF / 16, NN / 128, BB), 256, 0, stream>>>(A, h1p, dvec, b1, x1);
    // h2' = D (X1 w2)   (reuses h1p buffer)
    h2_kernel<<<dim3(FF / 16, NN / 128, BB), 256, 0, stream>>>(x1, w2, dvec, h1p);
    // out = (D A h2' + b2)^T + v
    spmm_out_kernel<<<dim3(FF / 16, NN / 128, BB), 256, 0, stream>>>(A, h1p, dvec, b2, v, out);
}